// PAM_Module_335007449965
// MI455X (gfx1250) — compile-verified
//
#include <hip/hip_runtime.h>
#include <hip/hip_bf16.h>

// ---------------------------------------------------------------------------
// PAM (linear-attention) module for MI455X (gfx1250), wave32 + WMMA f32.
// B=8, C=256, d=32, N=16384. Bandwidth-bound (~0.4GB traffic @ 23.3TB/s),
// so we use full-precision V_WMMA_F32_16X16X4_F32 throughout.
// ---------------------------------------------------------------------------

typedef __attribute__((ext_vector_type(2))) float v2f;
typedef __attribute__((ext_vector_type(8))) float v8f;

#define BATCH 8
#define CH    256
#define DD    32
#define NN    16384
#define NT    64          // n-tile columns per inner iteration
#define CHUNK 256         // n columns per workgroup
#define NCHUNKS 64        // NN / CHUNK

#define XS_STR  68        // 64 + 4 pad (bank spread)
#define W_STR   260       // 256 + 4 pad
#define KVT_STR 36        // 32 + 4 pad

__device__ __forceinline__ float softplus_f(float z) {
  // numerically stable softplus, matches jax.nn.softplus
  return z > 0.f ? z + log1pf(expf(-z)) : log1pf(expf(z));
}

// A-matrix 16x4 fragment (ISA 7.12.2): lanes 0-15 row M=lane, VGPR0=K+0,
// VGPR1=K+1; lanes 16-31 row M=lane-16, VGPR0=K+2, VGPR1=K+3.
// Element A[m][k] stored row-major at s[m*stride + k].
__device__ __forceinline__ v2f load_a(const float* s, int stride, int mBase, int kBase) {
  const int l = threadIdx.x & 31;
  const int m = mBase + (l & 15);
  const int k = kBase + ((l >> 4) << 1);
  v2f a;
  a.x = s[m * stride + k];
  a.y = s[m * stride + k + 1];
  return a;
}

// B-matrix 4x16 fragment: one row striped across lanes within one VGPR
// (mirrors A striping): lanes 0-15 col N=lane, VGPR0=K+0, VGPR1=K+1;
// lanes 16-31 col N=lane-16, VGPR0=K+2, VGPR1=K+3.
// Element B[k][n] stored at s[k*stride + n].
__device__ __forceinline__ v2f load_b(const float* s, int stride, int kBase, int nBase) {
  const int l = threadIdx.x & 31;
  const int n = nBase + (l & 15);
  const int k = kBase + ((l >> 4) << 1);
  v2f b;
  b.x = s[k * stride + n];
  b.y = s[(k + 1) * stride + n];
  return b;
}

// Same as load_b but the logical B[k][col] is stored transposed: s[col*stride + k].
__device__ __forceinline__ v2f load_bT(const float* s, int stride, int kBase, int colBase) {
  const int l = threadIdx.x & 31;
  const int n = colBase + (l & 15);
  const int k = kBase + ((l >> 4) << 1);
  v2f b;
  b.x = s[n * stride + k];
  b.y = s[n * stride + k + 1];
  return b;
}

__device__ __forceinline__ v8f wmma4(v2f a, v2f b, v8f c) {
  // D = A(16x4) * B(4x16) + C, full f32
  return __builtin_amdgcn_wmma_f32_16x16x4_f32(
      /*neg_a=*/false, a, /*neg_b=*/false, b,
      /*c_mod=*/(short)0, c, /*reuse_a=*/false, /*reuse_b=*/false);
}

// ---------------------------------------------------------------------------
__global__ void pam_zero(float* __restrict__ p, int n) {
  int i = blockIdx.x * 256 + threadIdx.x;
  if (i < n) p[i] = 0.f;
}

// ---------------------------------------------------------------------------
// Phase 1: K = softplus(wk x + bk); KX += K x^T ; Ksum += row-sum(K).
// grid = BATCH * NCHUNKS blocks, 256 threads (8 waves).
__global__ __launch_bounds__(256) void pam_phase1(
    const float* __restrict__ x, const float* __restrict__ wk,
    const float* __restrict__ bk, float* __restrict__ KX,
    float* __restrict__ Ksum) {
  __shared__ __align__(16) float wks[DD * W_STR];
  __shared__ __align__(16) float xs[CH * XS_STR];
  __shared__ __align__(16) float ks[DD * XS_STR];
  __shared__ float bks[DD];

  const int tid  = threadIdx.x;
  const int lane = tid & 31;
  const int wave = tid >> 5;
  const int b     = blockIdx.x >> 6;   // NCHUNKS = 64
  const int chunk = blockIdx.x & 63;
  const long nBase = (long)chunk * CHUNK;
  const float* xb = x + (long)b * CH * NN;

  // stage wk (32x256) and bk
  for (int i = tid; i < DD * CH; i += 256) {
    int m = i >> 8, c = i & 255;
    wks[m * W_STR + c] = wk[i];
  }
  if (tid < DD) bks[tid] = bk[tid];

  float ksum_acc = 0.f;
  v8f accKX[4];
  v8f vzero = {};
#pragma unroll
  for (int i = 0; i < 4; ++i) accKX[i] = vzero;

  const int mt  = wave & 1;   // which 16 rows of d
  const int grp = wave >> 1;  // 0..3

  for (int t = 0; t < CHUNK / NT; ++t) {
    const long n0 = nBase + (long)t * NT;
    __syncthreads();
    // stage x tile [256 x 64], float4 coalesced
#pragma unroll
    for (int i = 0; i < 16; ++i) {
      int linear = tid + 256 * i;   // 0..4095 float4 slots
      int c  = linear >> 4;
      int j4 = linear & 15;
      float4 v = *(const float4*)(xb + (long)c * NN + n0 + j4 * 4);
      *(float4*)(xs + c * XS_STR + j4 * 4) = v;
    }
    __syncthreads();

    // K projection: wave computes D-tile (rows mt*16.., cols grp*16..), K-dim = C
    {
      v8f acc = vzero;
      for (int c0 = 0; c0 < CH; c0 += 4) {
        v2f a  = load_a(wks, W_STR, mt * 16, c0);
        v2f bb = load_b(xs, XS_STR, c0, grp * 16);
        acc = wmma4(a, bb, acc);
      }
      // bias + softplus, write K tile to LDS (D-fragment layout)
      const int mloc = (lane < 16) ? 0 : 8;
      const int nl = grp * 16 + (lane & 15);
#pragma unroll
      for (int r = 0; r < 8; ++r) {
        int m = mt * 16 + r + mloc;
        ks[m * XS_STR + nl] = softplus_f(acc[r] + bks[m]);
      }
    }
    __syncthreads();

    // Ksum accumulation (wave 0)
    if (tid < DD) {
      float s = 0.f;
      for (int j = 0; j < NT; ++j) s += ks[tid * XS_STR + j];
      ksum_acc += s;
    }

    // KX accumulation: KX[m, e] += sum_n K[m,n] * x[e,n]
    // wave owns m rows mt*16.. and e-tiles grp*64 .. grp*64+63
#pragma unroll
    for (int i = 0; i < 4; ++i) {
      const int e0 = grp * 64 + i * 16;
      for (int k0 = 0; k0 < NT; k0 += 4) {
        v2f a  = load_a(ks, XS_STR, mt * 16, k0);
        v2f bb = load_bT(xs, XS_STR, k0, e0);  // B[k=n][e] = xs[e][n]
        accKX[i] = wmma4(a, bb, accKX[i]);
      }
    }
  }

  // flush partials
  if (tid < DD) atomicAdd(&Ksum[b * DD + tid], ksum_acc);
  {
    const int mloc = (lane < 16) ? 0 : 8;
    const int el = lane & 15;
#pragma unroll
    for (int i = 0; i < 4; ++i) {
      int e = grp * 64 + i * 16 + el;
#pragma unroll
      for (int r = 0; r < 8; ++r) {
        int m = mt * 16 + r + mloc;
        atomicAdd(&KX[(b * DD + m) * CH + e], accKX[i][r]);
      }
    }
  }
}

// ---------------------------------------------------------------------------
// Phase 2 (tiny): KVt[b][c][m] = sum_e KX[b][m][e]*wv[c][e] + bv[c]*Ksum[b][m]
//                 KsumEps = Ksum + 1e-6
__global__ __launch_bounds__(256) void pam_phase2(
    const float* __restrict__ KX, const float* __restrict__ Ksum,
    const float* __restrict__ wv, const float* __restrict__ bv,
    float* __restrict__ KVt, float* __restrict__ KsumEps) {
  __shared__ float kxs[DD * CH];
  __shared__ float ks0[DD];
  const int b = blockIdx.x;
  const int tid = threadIdx.x;
  for (int i = tid; i < DD * CH; i += 256) kxs[i] = KX[b * DD * CH + i];
  if (tid < DD) ks0[tid] = Ksum[b * DD + tid];
  __syncthreads();

  const int c = tid;
  float acc[DD];
#pragma unroll
  for (int m = 0; m < DD; ++m) acc[m] = 0.f;
  const float* wvr = wv + (long)c * CH;
  for (int e = 0; e < CH; e += 4) {
    float4 w4 = *(const float4*)(wvr + e);
#pragma unroll
    for (int m = 0; m < DD; ++m) {
      acc[m] += w4.x * kxs[m * CH + e]     + w4.y * kxs[m * CH + e + 1] +
                w4.z * kxs[m * CH + e + 2] + w4.w * kxs[m * CH + e + 3];
    }
  }
  const float bvc = bv[c];
#pragma unroll
  for (int m = 0; m < DD; ++m)
    KVt[((long)b * CH + c) * DD + m] = acc[m] + bvc * ks0[m];
  if (tid < DD) KsumEps[b * DD + tid] = ks0[tid] + 1e-6f;
}

// ---------------------------------------------------------------------------
// Phase 3: Q^T = softplus(wq x + bq); norm = 1/(Q^T . KsumEps);
//          out = x + gamma * norm * (KVt * Q^T)
__global__ __launch_bounds__(256) void pam_phase3(
    const float* __restrict__ x, const float* __restrict__ wq,
    const float* __restrict__ bq, const float* __restrict__ KVt,
    const float* __restrict__ KsumEps, const float* __restrict__ gamma,
    float* __restrict__ out) {
  __shared__ __align__(16) float wqs[DD * W_STR];
  __shared__ __align__(16) float xs[CH * XS_STR];
  __shared__ __align__(16) float qs[DD * XS_STR];
  __shared__ __align__(16) float kvts[CH * KVT_STR];
  __shared__ float bqs[DD];
  __shared__ float kses[DD];
  __shared__ float norms[NT];

  const int tid  = threadIdx.x;
  const int lane = tid & 31;
  const int wave = tid >> 5;
  const int b     = blockIdx.x >> 6;
  const int chunk = blockIdx.x & 63;
  const long nBase = (long)chunk * CHUNK;
  const float* xb = x + (long)b * CH * NN;
  float* ob = out + (long)b * CH * NN;
  const float g = gamma[0];

  for (int i = tid; i < DD * CH; i += 256) {
    int m = i >> 8, c = i & 255;
    wqs[m * W_STR + c] = wq[i];
  }
  for (int i = tid; i < CH * DD; i += 256) {
    int c = i >> 5, m = i & 31;
    kvts[c * KVT_STR + m] = KVt[(long)b * CH * DD + i];
  }
  if (tid < DD) {
    bqs[tid]  = bq[tid];
    kses[tid] = KsumEps[b * DD + tid];
  }

  const int mt  = wave & 1;
  const int grp = wave >> 1;
  v8f vzero = {};

  for (int t = 0; t < CHUNK / NT; ++t) {
    const long n0 = nBase + (long)t * NT;
    __syncthreads();
#pragma unroll
    for (int i = 0; i < 16; ++i) {
      int linear = tid + 256 * i;
      int c  = linear >> 4;
      int j4 = linear & 15;
      float4 v = *(const float4*)(xb + (long)c * NN + n0 + j4 * 4);
      *(float4*)(xs + c * XS_STR + j4 * 4) = v;
    }
    __syncthreads();

    // Q^T tile [32 x 64] via WMMA (same structure as K projection)
    {
      v8f acc = vzero;
      for (int c0 = 0; c0 < CH; c0 += 4) {
        v2f a  = load_a(wqs, W_STR, mt * 16, c0);
        v2f bb = load_b(xs, XS_STR, c0, grp * 16);
        acc = wmma4(a, bb, acc);
      }
      const int mloc = (lane < 16) ? 0 : 8;
      const int nl = grp * 16 + (lane & 15);
#pragma unroll
      for (int r = 0; r < 8; ++r) {
        int m = mt * 16 + r + mloc;
        qs[m * XS_STR + nl] = softplus_f(acc[r] + bqs[m]);
      }
    }
    __syncthreads();

    // norm[n] = 1 / sum_m Q^T[m,n] * (Ksum[m] + eps)
    if (tid < NT) {
      float s = 0.f;
#pragma unroll
      for (int m = 0; m < DD; ++m) s += qs[m * XS_STR + tid] * kses[m];
      norms[tid] = 1.f / s;
    }
    __syncthreads();

    // out tile: O[c, n] = sum_m KVt[c, m] * Q^T[m, n]; then residual+scale.
    // wave owns c-tiles {2*wave, 2*wave+1} x 4 n-subtiles
#pragma unroll
    for (int cti = 0; cti < 2; ++cti) {
      const int ct = wave * 2 + cti;
#pragma unroll
      for (int ntt = 0; ntt < 4; ++ntt) {
        v8f acc = vzero;
#pragma unroll
        for (int m0 = 0; m0 < DD; m0 += 4) {
          v2f a  = load_a(kvts, KVT_STR, ct * 16, m0);
          v2f bb = load_b(qs, XS_STR, m0, ntt * 16);
          acc = wmma4(a, bb, acc);
        }
        const int mloc = (lane < 16) ? 0 : 8;
        const int nl = ntt * 16 + (lane & 15);
        const float sc = norms[nl] * g;
        const long n = n0 + nl;
#pragma unroll
        for (int r = 0; r < 8; ++r) {
          int c = ct * 16 + r + mloc;
          ob[(long)c * NN + n] = xs[c * XS_STR + nl] + sc * acc[r];
        }
      }
    }
  }
}

// ---------------------------------------------------------------------------
extern "C" void kernel_launch(void* const* d_in, const int* in_sizes, int n_in,
                              void* d_out, int out_size, void* d_ws, size_t ws_size,
                              hipStream_t stream) {
  const float* x     = (const float*)d_in[0];
  const float* wq    = (const float*)d_in[1];
  const float* bq    = (const float*)d_in[2];
  const float* wk    = (const float*)d_in[3];
  const float* bk    = (const float*)d_in[4];
  const float* wv    = (const float*)d_in[5];
  const float* bv    = (const float*)d_in[6];
  const float* gamma = (const float*)d_in[7];
  float* out = (float*)d_out;

  float* ws = (float*)d_ws;
  float* KX      = ws;                       // B*D*C = 65536 floats
  float* Ksum    = ws + BATCH * DD * CH;     // B*D   = 256
  float* KVt     = Ksum + BATCH * DD;        // B*C*D = 65536
  float* KsumEps = KVt + BATCH * CH * DD;    // B*D   = 256

  const int nzero = BATCH * DD * CH + BATCH * DD;  // KX + Ksum contiguous
  pam_zero<<<(nzero + 255) / 256, 256, 0, stream>>>(KX, nzero);
  pam_phase1<<<BATCH * NCHUNKS, 256, 0, stream>>>(x, wk, bk, KX, Ksum);
  pam_phase2<<<BATCH, 256, 0, stream>>>(KX, Ksum, wv, bv, KVt, KsumEps);
  pam_phase3<<<BATCH * NCHUNKS, 256, 0, stream>>>(x, wq, bq, KVt, KsumEps, gamma, out);
}